// Memory_68135361183929
// MI455X (gfx1250) — compile-verified
//
#include <hip/hip_runtime.h>
#include <math.h>

#define B_ 256
#define N_ 2048
#define M_ 128
#define EPS_ 1e-8f

typedef float v2f __attribute__((ext_vector_type(2)));
typedef float v4f __attribute__((ext_vector_type(4)));
typedef float v8f __attribute__((ext_vector_type(8)));

// ---------------------------------------------------------------------------
// Kernel 1: dot[b,n] = <k[b], memory[b,n,:]> via V_WMMA_F32_16X16X4_F32,
//           rnorm[b,n] = ||memory[b,n,:]||  (single streaming pass over memory)
// One wave (32 lanes) computes a 16-row tile; 32 WMMA issues cover M=128 (K=4 each).
// A layout (16x4 f32): lanes 0-15 hold K=0,1 ; lanes 16-31 hold K=2,3.
// B layout (4x16 f32): lanes 0-15 = rows K0/K1 ; lanes 16-31 = rows K2/K3;
//   k-chunk replicated across all 16 columns -> every column of D is the dot.
// ---------------------------------------------------------------------------
__global__ __launch_bounds__(256) void ntm_dot_norm_kernel(
    const float* __restrict__ memory, const float* __restrict__ k,
    float* __restrict__ dot, float* __restrict__ rnorm)
{
    __shared__ float ldsK[M_];
    const int b   = blockIdx.x >> 4;           // 16 blocks per batch
    const int tid = threadIdx.x;
    if (tid < M_) ldsK[tid] = k[(size_t)b * M_ + tid];
    __syncthreads();

    const int wave = tid >> 5;
    const int lane = tid & 31;
    const int n0   = (((blockIdx.x & 15) << 3) + wave) << 4;  // tile base row
    const int row  = lane & 15;
    const int half = (lane >> 4) << 1;          // 0 for lanes 0-15, 2 for 16-31

    const float* mrow = memory + ((size_t)b * N_ + (n0 + row)) * M_ + half;

    v8f c = {0.f, 0.f, 0.f, 0.f, 0.f, 0.f, 0.f, 0.f};
    float ss = 0.f;
    #pragma unroll
    for (int kc = 0; kc < 32; ++kc) {
        v2f av = *reinterpret_cast<const v2f*>(mrow + kc * 4);
        v2f bv;
        bv.x = ldsK[kc * 4 + half];
        bv.y = ldsK[kc * 4 + half + 1];
        ss += av.x * av.x + av.y * av.y;
        c = __builtin_amdgcn_wmma_f32_16x16x4_f32(
                /*neg_a=*/false, av, /*neg_b=*/false, bv,
                /*c_mod=*/(short)0, c, /*reuse_a=*/false, /*reuse_b=*/false);
    }
    // lanes (L, L+16) hold the two halves of row L's sum of squares
    float ssFull = ss + __shfl_xor(ss, 16);
    if (lane < 16) rnorm[(size_t)b * N_ + n0 + lane] = sqrtf(ssFull);

    // All D columns identical: lane 0 has rows 0-7 in c[0..7], lane 16 rows 8-15.
    if (lane == 0 || lane == 16) {
        float* dp = dot + (size_t)b * N_ + n0 + ((lane >> 4) << 3);
        #pragma unroll
        for (int j = 0; j < 8; ++j) dp[j] = c[j];
    }
}

// ---------------------------------------------------------------------------
// Kernel 2: per-batch weight pipeline. One block (256 thr) per batch.
// cosine -> softmax(beta*sim) -> gate -> circular 3-tap shift -> pow(y) -> norm
// ---------------------------------------------------------------------------
__global__ __launch_bounds__(256) void ntm_weights_kernel(
    const float* __restrict__ dot, const float* __restrict__ rnorm,
    const float* __restrict__ k, const float* __restrict__ prev_w,
    const float* __restrict__ beta, const float* __restrict__ g,
    const float* __restrict__ s, const float* __restrict__ y,
    float* __restrict__ out_w, float* __restrict__ out_read)
{
    __shared__ float wgLds[N_];
    __shared__ float red[256];
    const int b = blockIdx.x;
    const int t = threadIdx.x;

    // ||k[b]||
    float kv = 0.f;
    if (t < M_) { float v = k[(size_t)b * M_ + t]; kv = v * v; }
    red[t] = kv;
    __syncthreads();
    for (int off = 128; off > 0; off >>= 1) {
        if (t < off) red[t] += red[t + off];
        __syncthreads();
    }
    const float knorm = sqrtf(red[0]);
    __syncthreads();

    const float betaB = beta[b];
    const float gB    = g[b];
    const float yB    = y[b];
    const float s0 = s[b * 3 + 0], s1 = s[b * 3 + 1], s2 = s[b * 3 + 2];

    float x[8];
    #pragma unroll
    for (int i = 0; i < 8; ++i) {
        const int n = t + (i << 8);
        const float denom = fmaxf(knorm * rnorm[(size_t)b * N_ + n], EPS_);
        x[i] = betaB * (dot[(size_t)b * N_ + n] / denom);
    }
    // stable softmax: max reduce
    float lmax = x[0];
    #pragma unroll
    for (int i = 1; i < 8; ++i) lmax = fmaxf(lmax, x[i]);
    red[t] = lmax;
    __syncthreads();
    for (int off = 128; off > 0; off >>= 1) {
        if (t < off) red[t] = fmaxf(red[t], red[t + off]);
        __syncthreads();
    }
    const float xmax = red[0];
    __syncthreads();

    float ex[8]; float lsum = 0.f;
    #pragma unroll
    for (int i = 0; i < 8; ++i) { ex[i] = expf(x[i] - xmax); lsum += ex[i]; }
    red[t] = lsum;
    __syncthreads();
    for (int off = 128; off > 0; off >>= 1) {
        if (t < off) red[t] += red[t + off];
        __syncthreads();
    }
    const float esum = red[0];
    __syncthreads();

    // gate + stage into LDS ring for the circular shift
    #pragma unroll
    for (int i = 0; i < 8; ++i) {
        const int n = t + (i << 8);
        const float wc = ex[i] / esum;
        wgLds[n] = gB * wc + (1.f - gB) * prev_w[(size_t)b * N_ + n];
    }
    __syncthreads();

    // wt[n] = wg[n-1]*s0 + wg[n]*s1 + wg[n+1]*s2 (circular), then sharpen
    float p[8]; float psumL = 0.f;
    #pragma unroll
    for (int i = 0; i < 8; ++i) {
        const int n = t + (i << 8);
        float wt = wgLds[(n + N_ - 1) & (N_ - 1)] * s0
                 + wgLds[n] * s1
                 + wgLds[(n + 1) & (N_ - 1)] * s2;
        wt = fmaxf(wt, 0.f);
        p[i] = powf(wt, yB);
        psumL += p[i];
    }
    red[t] = psumL;
    __syncthreads();
    for (int off = 128; off > 0; off >>= 1) {
        if (t < off) red[t] += red[t + off];
        __syncthreads();
    }
    const float psum = red[0];

    #pragma unroll
    for (int i = 0; i < 8; ++i) {
        const int n = t + (i << 8);
        out_w[(size_t)b * N_ + n] = p[i] / psum;
    }
    // zero the read slice of d_out (poisoned by harness; K3 accumulates into it)
    if (t < M_) out_read[(size_t)b * M_ + t] = 0.f;
}

// ---------------------------------------------------------------------------
// Kernel 3: fused read + erase/add memory update (second streaming pass).
// 8 blocks per batch, 256 rows each; b128 loads/stores, LDS partial-read
// reduction, global f32 atomics to finish the cross-block read reduction.
// ---------------------------------------------------------------------------
__global__ __launch_bounds__(256) void ntm_readwrite_kernel(
    const float* __restrict__ memory, const float* __restrict__ w,
    const float* __restrict__ e, const float* __restrict__ a,
    float* __restrict__ new_memory, float* __restrict__ out_read)
{
    __shared__ float rbuf[8][M_];
    const int b    = blockIdx.x >> 3;
    const int n0   = (blockIdx.x & 7) << 8;     // 256 rows per block
    const int t    = threadIdx.x;
    const int mlan = t & 31;
    const int rgrp = t >> 5;
    const int m4   = mlan << 2;

    const v4f e4 = *reinterpret_cast<const v4f*>(e + (size_t)b * M_ + m4);
    const v4f a4 = *reinterpret_cast<const v4f*>(a + (size_t)b * M_ + m4);
    v4f racc = {0.f, 0.f, 0.f, 0.f};

    for (int r = rgrp; r < 256; r += 8) {
        const int n = n0 + r;
        const float wn = w[(size_t)b * N_ + n];
        const float* mp = memory + ((size_t)b * N_ + n) * M_ + m4;
        __builtin_prefetch(mp + 8 * M_, 0, 0);   // next row for this thread
        const v4f mv = *reinterpret_cast<const v4f*>(mp);
        const v4f nm = mv * (1.f - wn * e4) + wn * a4;
        *reinterpret_cast<v4f*>(new_memory + ((size_t)b * N_ + n) * M_ + m4) = nm;
        racc += wn * mv;
    }
    #pragma unroll
    for (int j = 0; j < 4; ++j) rbuf[rgrp][m4 + j] = racc[j];
    __syncthreads();
    if (rgrp == 0) {
        #pragma unroll
        for (int j = 0; j < 4; ++j) {
            float sum = 0.f;
            #pragma unroll
            for (int rr = 0; rr < 8; ++rr) sum += rbuf[rr][m4 + j];
            atomicAdd(&out_read[(size_t)b * M_ + m4 + j], sum);
        }
    }
}

// ---------------------------------------------------------------------------
extern "C" void kernel_launch(void* const* d_in, const int* in_sizes, int n_in,
                              void* d_out, int out_size, void* d_ws, size_t ws_size,
                              hipStream_t stream) {
    (void)in_sizes; (void)n_in; (void)out_size; (void)ws_size;

    const float* memory = (const float*)d_in[0];
    const float* prev_w = (const float*)d_in[1];
    const float* k      = (const float*)d_in[2];
    const float* beta   = (const float*)d_in[3];
    const float* g      = (const float*)d_in[4];
    const float* s      = (const float*)d_in[5];
    const float* y      = (const float*)d_in[6];
    const float* e      = (const float*)d_in[7];
    const float* a      = (const float*)d_in[8];

    float* out_w      = (float*)d_out;                       // B*N
    float* out_read   = out_w + (size_t)B_ * N_;             // B*M
    float* out_newmem = out_read + (size_t)B_ * M_;          // B*N*M

    float* ws_dot   = (float*)d_ws;                          // B*N
    float* ws_rnorm = ws_dot + (size_t)B_ * N_;              // B*N  (4 MB total)

    // K1: 16 blocks/batch * 256 batches; 8 waves/block, 16-row WMMA tile/wave
    ntm_dot_norm_kernel<<<B_ * 16, 256, 0, stream>>>(memory, k, ws_dot, ws_rnorm);
    // K2: one block per batch
    ntm_weights_kernel<<<B_, 256, 0, stream>>>(ws_dot, ws_rnorm, k, prev_w,
                                               beta, g, s, y, out_w, out_read);
    // K3: 8 blocks per batch
    ntm_readwrite_kernel<<<B_ * 8, 256, 0, stream>>>(memory, out_w, e, a,
                                                     out_newmem, out_read);
}